// HIPModule_74586402062653
// MI455X (gfx1250) — compile-verified
//
#include <hip/hip_runtime.h>
#include <hip/hip_bf16.h>

// MI455X / gfx1250 fused ANI-style network.
// fp32 matrix path: V_WMMA_F32_16X16X4_F32 (wave32, 16x16 D tile, K=4/instr).

#define FDIM 80
#define ST   84          // LDS row stride (floats): 84*4B = 336B, 16B aligned, conflict-free
#define NT5  5           // 80 = 5 N-tiles of 16

typedef __attribute__((ext_vector_type(2))) float v2f;
typedef __attribute__((ext_vector_type(8))) float v8f;

__device__ __forceinline__ float softplus_f(float x) {
  // max(x,0) + log(1 + exp(-|x|)) via the hardware TRANS pipe
  // (v_exp_f32 / v_log_f32). Absolute error ~1e-7, dominated by fmax term
  // exactly where the log1p shortcut is least precise.
  return fmaxf(x, 0.0f) + __logf(1.0f + __expf(-fabsf(x)));
}

__device__ __forceinline__ v8f wmma4(v2f a, v2f b, v8f c) {
  // D = A(16x4) * B(4x16) + C(16x16), all f32
  return __builtin_amdgcn_wmma_f32_16x16x4_f32(
      /*neg_a=*/false, a, /*neg_b=*/false, b,
      /*c_mod=*/(short)0, c, /*reuse_a=*/false, /*reuse_b=*/false);
}

// ---------------------------------------------------------------------------
// Kernel 1: pair message + scatter.
// 64 pairs per WG -> 128 gathered rows. 8 waves, each owns one 16-row subtile.
// fmsg = sp(sp(f12) @ WJ + bJ); pair_msg = aev_term * fmsg; atomic scatter.
// ---------------------------------------------------------------------------
__global__ __launch_bounds__(256) void pair_scatter_kernel(
    const float* __restrict__ feat, const float* __restrict__ radial,
    const int* __restrict__ idx12,
    const float* __restrict__ Wg, const float* __restrict__ bg,
    const float* __restrict__ WJ, const float* __restrict__ bJ,
    float* __restrict__ proto, int P)
{
  __shared__ float sX[128 * ST];   // sp(features[gather]) rows
  __shared__ float sA[64 * ST];    // aev_term per pair
  __shared__ int   sIdx[128];

  const int tid  = threadIdx.x;
  const int lane = tid & 31, wave = tid >> 5;
  const int half = lane >> 4, l15 = lane & 15;
  const int mrow = wave * 16;
  const int p0   = blockIdx.x * 64;

  if (tid < 128) {
    const int side = tid >> 6, j = tid & 63;
    sIdx[tid] = idx12[side * P + p0 + j];
  }
  __syncthreads();

  // gather + softplus into LDS (L2-resident feature table)
  for (int i = tid; i < 128 * FDIM; i += 256) {
    const int r = i / FDIM, c = i - r * FDIM;
    sX[r * ST + c] = softplus_f(feat[(size_t)sIdx[r] * FDIM + c]);
  }

  // aev_term = radial @ Wg + bg  (K=20: cheap VALU, weights hot in L0/L2)
  {
    const int j = tid >> 2, fb = (tid & 3) * 20;
    float rrow[20];
    #pragma unroll
    for (int k = 0; k < 20; ++k) rrow[k] = radial[(size_t)(p0 + j) * 20 + k];
    #pragma unroll 1
    for (int f = 0; f < 20; ++f) {
      float s = bg[fb + f];
      #pragma unroll
      for (int k = 0; k < 20; ++k) s += rrow[k] * Wg[k * FDIM + fb + f];
      sA[j * ST + fb + f] = s;
    }
  }
  __syncthreads();

  v8f acc[NT5];
  const v8f z8 = {0, 0, 0, 0, 0, 0, 0, 0};
  #pragma unroll
  for (int nt = 0; nt < NT5; ++nt) acc[nt] = z8;

  __builtin_prefetch(WJ, 0, 1);
  #pragma unroll 1
  for (int kk = 0; kk < FDIM; kk += 4) {
    const float* ap = &sX[(mrow + l15) * ST + kk + 2 * half];
    v2f a; a.x = ap[0]; a.y = ap[1];
    const float* wp = &WJ[(kk + 2 * half) * FDIM + l15];
    #pragma unroll
    for (int nt = 0; nt < NT5; ++nt) {
      v2f b; b.x = wp[nt * 16]; b.y = wp[FDIM + nt * 16];
      acc[nt] = wmma4(a, b, acc[nt]);
    }
  }

  // epilogue: softplus, gate by aev, atomic scatter-add (hw f32 atomics)
  #pragma unroll
  for (int nt = 0; nt < NT5; ++nt) {
    const int n  = nt * 16 + l15;
    const float bn = bJ[n];
    #pragma unroll
    for (int r = 0; r < 8; ++r) {
      const int Lr = mrow + r + 8 * half;          // 0..127
      const float v = softplus_f(acc[nt][r] + bn) * sA[(Lr & 63) * ST + n];
      unsafeAtomicAdd(&proto[(size_t)sIdx[Lr] * FDIM + n], v);
    }
  }
}

// ---------------------------------------------------------------------------
// Kernel 2: fully fused per-atom MLP chain (20 x [80x80] GEMMs + GEMV).
// 64 atoms per WG, 4 waves; activation tile resident in LDS end-to-end.
// Each wave only touches its own 16 rows -> one barrier total.
// ---------------------------------------------------------------------------
__global__ __launch_bounds__(128) void atom_mlp_kernel(
    const int* __restrict__ species, const float* __restrict__ feat,
    const float* __restrict__ proto,
    const float* __restrict__ WI,   const float* __restrict__ bI,
    const float* __restrict__ iW1,  const float* __restrict__ ib1,
    const float* __restrict__ iW2,  const float* __restrict__ ib2,
    const float* __restrict__ gate, const float* __restrict__ Wint,
    const float* __restrict__ bint,
    const float* __restrict__ aW1,  const float* __restrict__ ab1,
    const float* __restrict__ aW2,  const float* __restrict__ ab2,
    const float* __restrict__ oW1,  const float* __restrict__ ob1,
    const float* __restrict__ oW2,  const float* __restrict__ ob2,
    const float* __restrict__ Wout, const float* __restrict__ bout,
    float* __restrict__ out_e, float* __restrict__ out_f)
{
  __shared__ float bufA[64 * ST];
  __shared__ float bufB[64 * ST];

  const int tid  = threadIdx.x;
  const int lane = tid & 31, wave = tid >> 5;
  const int half = lane >> 4, l15 = lane & 15;
  const int mrow = wave * 16;
  const int row0 = blockIdx.x * 64;

  for (int i = tid; i < 64 * FDIM; i += 128) {
    const int r = i / FDIM, c = i - r * FDIM;
    bufA[r * ST + c] = feat[(size_t)(row0 + r) * FDIM + c];
  }
  __syncthreads();     // only cross-wave sync needed in this kernel

  v8f acc[NT5];
  const v8f z8 = {0, 0, 0, 0, 0, 0, 0, 0};

  auto zacc = [&]() {
    #pragma unroll
    for (int nt = 0; nt < NT5; ++nt) acc[nt] = z8;
  };

  // acc += src(16x80, this wave's rows) @ W(80x80 row-major, from L2)
  auto gemmA = [&](const float* src, const float* __restrict__ W) {
    __builtin_prefetch(W, 0, 1);
    #pragma unroll 1
    for (int kk = 0; kk < FDIM; kk += 4) {
      const float* ap = &src[(mrow + l15) * ST + kk + 2 * half];
      v2f a; a.x = ap[0]; a.y = ap[1];
      const float* wp = &W[(kk + 2 * half) * FDIM + l15];
      #pragma unroll
      for (int nt = 0; nt < NT5; ++nt) {
        v2f b; b.x = wp[nt * 16]; b.y = wp[FDIM + nt * 16];
        acc[nt] = wmma4(a, b, acc[nt]);
      }
    }
  };

  // iterate the D-fragment positions (M = r + 8*half, N = nt*16 + l15)
  auto forD = [&](auto&& fn) {
    #pragma unroll
    for (int nt = 0; nt < NT5; ++nt) {
      const int n = nt * 16 + l15;
      #pragma unroll
      for (int r = 0; r < 8; ++r) {
        const int Lr = mrow + r + 8 * half;
        fn(nt, r, Lr, n);
      }
    }
  };

  // x0 = proto_scatter + (feat @ WI + bI) * mask
  zacc();
  gemmA(bufA, WI);
  forD([&](int nt, int r, int Lr, int n) {
    const int g = row0 + Lr;
    const float msk = (species[g] != -1) ? 1.f : 0.f;
    bufA[Lr * ST + n] = proto[(size_t)g * FDIM + n] + (acc[nt][r] + bI[n]) * msk;
  });

  auto resblock = [&](const float* W1, const float* b1v,
                      const float* W2, const float* b2v) {
    zacc();
    gemmA(bufA, W1);
    forD([&](int nt, int r, int Lr, int n) {
      bufB[Lr * ST + n] = softplus_f(acc[nt][r] + b1v[n]);
    });
    zacc();
    gemmA(bufB, W2);
    forD([&](int nt, int r, int Lr, int n) {   // residual lands on D layout
      bufA[Lr * ST + n] = softplus_f(acc[nt][r] + b2v[n] + bufA[Lr * ST + n]);
    });
  };

  #pragma unroll 1
  for (int b = 0; b < 3; ++b)
    resblock(iW1 + b * FDIM * FDIM, ib1 + b * FDIM,
             iW2 + b * FDIM * FDIM, ib2 + b * FDIM);

  // nd = feat*gate + sp(message) @ Wint + bint
  for (int i = lane; i < 16 * FDIM; i += 32) {
    const int r = i / FDIM, c = i - r * FDIM;
    bufB[(mrow + r) * ST + c] = softplus_f(bufA[(mrow + r) * ST + c]);
  }
  zacc();
  gemmA(bufB, Wint);
  forD([&](int nt, int r, int Lr, int n) {
    const int g = row0 + Lr;
    bufA[Lr * ST + n] = feat[(size_t)g * FDIM + n] * gate[n] + acc[nt][r] + bint[n];
  });

  #pragma unroll 1
  for (int b = 0; b < 3; ++b)
    resblock(aW1 + b * FDIM * FDIM, ab1 + b * FDIM,
             aW2 + b * FDIM * FDIM, ab2 + b * FDIM);

  // out_features = nd * mask
  forD([&](int nt, int r, int Lr, int n) {
    const int g = row0 + Lr;
    const float msk = (species[g] != -1) ? 1.f : 0.f;
    out_f[(size_t)g * FDIM + n] = bufA[Lr * ST + n] * msk;
  });

  #pragma unroll 1
  for (int b = 0; b < 3; ++b)
    resblock(oW1 + b * FDIM * FDIM, ob1 + b * FDIM,
             oW2 + b * FDIM * FDIM, ob2 + b * FDIM);

  // energies = (sp(y) @ Wout + bout) * mask
  if (lane < 16) {
    const int Lr = mrow + l15;
    const int g  = row0 + Lr;
    float e = bout[0];
    #pragma unroll 1
    for (int c = 0; c < FDIM; ++c)
      e += softplus_f(bufA[Lr * ST + c]) * Wout[c];
    const float msk = (species[g] != -1) ? 1.f : 0.f;
    out_e[g] = e * msk;
  }
}

// ---------------------------------------------------------------------------
extern "C" void kernel_launch(void* const* d_in, const int* in_sizes, int n_in,
                              void* d_out, int out_size, void* d_ws, size_t ws_size,
                              hipStream_t stream)
{
  const int*   species = (const int*)  d_in[0];
  const float* feat    = (const float*)d_in[1];
  const float* radial  = (const float*)d_in[2];
  const int*   idx12   = (const int*)  d_in[3];
  const float* WI   = (const float*)d_in[4];
  const float* bI   = (const float*)d_in[5];
  const float* Wg   = (const float*)d_in[6];
  const float* bg   = (const float*)d_in[7];
  const float* WJ   = (const float*)d_in[8];
  const float* bJ   = (const float*)d_in[9];
  const float* iW1  = (const float*)d_in[10];
  const float* ib1  = (const float*)d_in[11];
  const float* iW2  = (const float*)d_in[12];
  const float* ib2  = (const float*)d_in[13];
  const float* gate = (const float*)d_in[14];
  const float* Wint = (const float*)d_in[15];
  const float* bint = (const float*)d_in[16];
  const float* aW1  = (const float*)d_in[17];
  const float* ab1  = (const float*)d_in[18];
  const float* aW2  = (const float*)d_in[19];
  const float* ab2  = (const float*)d_in[20];
  const float* oW1  = (const float*)d_in[21];
  const float* ob1  = (const float*)d_in[22];
  const float* oW2  = (const float*)d_in[23];
  const float* ob2  = (const float*)d_in[24];
  const float* Wout = (const float*)d_in[25];
  const float* bout = (const float*)d_in[26];

  const int N = in_sizes[1] / FDIM;   // 16384
  const int P = in_sizes[2] / 20;     // 393216

  float* proto = (float*)d_ws;        // N*80 f32 scratch
  float* out_e = (float*)d_out;       // [N] energies first
  float* out_f = out_e + N;           // then [N,80] features

  (void)hipMemsetAsync(proto, 0, (size_t)N * FDIM * sizeof(float), stream);

  pair_scatter_kernel<<<P / 64, 256, 0, stream>>>(
      feat, radial, idx12, Wg, bg, WJ, bJ, proto, P);

  atom_mlp_kernel<<<N / 64, 128, 0, stream>>>(
      species, feat, proto,
      WI, bI, iW1, ib1, iW2, ib2, gate, Wint, bint,
      aW1, ab1, aW2, ab2, oW1, ob1, oW2, ob2, Wout, bout,
      out_e, out_f);
}